// ScaleLayer_14697378087297
// MI455X (gfx1250) — compile-verified
//
#include <hip/hip_runtime.h>

// Per-channel affine over NCHW f32: y[n,c,h,w] = x[n,c,h,w] * weight[c] + bias[c]
// x: (32, 64, 224, 224) f32  -> pure streaming, HBM-bandwidth bound.
//
// Design: 128-bit vector loads/stores with non-temporal hints (822 MB working
// set >> 192 MB L2), scalar (wave-uniform) weight/bias loads, 7x float4 per
// thread for MLP, plus a gfx1250 global_prefetch_b8 a block-stride ahead.

typedef float v4f __attribute__((ext_vector_type(4)));

#define HW_VEC   12544   // 224*224 / 4 float4s per channel plane
#define TPB      256
#define UNROLL   7       // 7 * 256 = 1792 float4 per block
#define BPP      7       // blocks per plane: 12544 / 1792 = 7 exactly

__global__ __launch_bounds__(TPB) void scale_nchw_f32_kernel(
    const float* __restrict__ x,
    const float* __restrict__ weight,
    const float* __restrict__ bias,
    float* __restrict__ y) {
  // blockIdx.y = n*C + c  (plane index); channel is scalar -> s_load for w/b.
  const int c = (int)(blockIdx.y & 63u);   // C == 64 per reference
  const float wc = weight[c];
  const float bc = bias[c];

  const v4f* __restrict__ xv = (const v4f*)x;
  v4f* __restrict__ yv = (v4f*)y;

  const size_t base = (size_t)blockIdx.y * HW_VEC
                    + (size_t)blockIdx.x * (TPB * UNROLL)
                    + threadIdx.x;

  // gfx1250 global_prefetch_b8: nudge the next block-stride of this stream
  // toward GL2. No counter impact; harmless if OOB (speculative, dropped).
  __builtin_prefetch(&xv[base + (size_t)(TPB * UNROLL)], 0, 0);

  // Issue all 7 independent 128-bit NT loads first (compiler batches them),
  // then FMA + NT store. th:NT keeps the one-touch stream out of L2 residency.
  v4f r[UNROLL];
#pragma unroll
  for (int i = 0; i < UNROLL; ++i) {
    r[i] = __builtin_nontemporal_load(&xv[base + (size_t)i * TPB]);
  }
#pragma unroll
  for (int i = 0; i < UNROLL; ++i) {
    v4f o = r[i] * wc + bc;   // componentwise fused multiply-add
    __builtin_nontemporal_store(o, &yv[base + (size_t)i * TPB]);
  }
}

extern "C" void kernel_launch(void* const* d_in, const int* in_sizes, int n_in,
                              void* d_out, int out_size, void* d_ws, size_t ws_size,
                              hipStream_t stream) {
  const float* x = (const float*)d_in[0];
  const float* w = (const float*)d_in[1];
  const float* b = (const float*)d_in[2];
  float* y = (float*)d_out;

  // planes = N*C; derived from input size so it tracks the reference shapes.
  const int planes = in_sizes[0] / (224 * 224);   // 32*64 = 2048
  dim3 grid(BPP, planes, 1);
  dim3 block(TPB, 1, 1);
  hipLaunchKernelGGL(scale_nchw_f32_kernel, grid, block, 0, stream, x, w, b, y);
}